// sLSTMblock_4415226380772
// MI455X (gfx1250) — compile-verified
//
#include <hip/hip_runtime.h>
#include <math.h>

// ---------------------------------------------------------------------------
// sLSTM block for MI455X (gfx1250), wave32 + V_WMMA_F32_16X16X4_F32.
// B=8, S=256, D=1024, DEPTH=4 (db=256), H=1365.
// Conv (~550 GFLOP fp32) -> WMMA via lag form  y[b,o,l]=sum_{i,t} xn[b,i,l-t]*Wrev[o,i,t],
// with async (ASYNCcnt) double-buffered LDS staging of xn rows and 16x64
// per-wave output strips for A-operand reuse.
// ---------------------------------------------------------------------------

typedef __attribute__((ext_vector_type(2))) float v2f;
typedef __attribute__((ext_vector_type(8))) float v8f;

#define Bc  8
#define Sc  256
#define Dc  1024
#define Hc  1365
#define ROWS (Bc * Sc)          // 2048

static __device__ __forceinline__ v8f wmma4(v2f a, v2f b, v8f c) {
  // (neg_a, A, neg_b, B, c_mod, C, reuse_a, reuse_b)
  return __builtin_amdgcn_wmma_f32_16x16x4_f32(false, a, false, b, (short)0, c,
                                               false, false);
}

static __device__ __forceinline__ float blockReduceSum(float v, float* sm) {
  const int t = threadIdx.x;
  sm[t] = v;
  __syncthreads();
#pragma unroll
  for (int off = 128; off > 0; off >>= 1) {
    if (t < off) sm[t] += sm[t + off];
    __syncthreads();
  }
  float r = sm[0];
  __syncthreads();
  return r;
}

// ------------------------- K1: row LayerNorm -------------------------------
__global__ void ln_kernel(const float* __restrict__ x, const float* __restrict__ g,
                          const float* __restrict__ b, float* __restrict__ y) {
  __shared__ float sm[256];
  const size_t off = (size_t)blockIdx.x * Dc;
  float v[4], s = 0.f, s2 = 0.f;
#pragma unroll
  for (int q = 0; q < 4; ++q) {
    float t = x[off + threadIdx.x + 256 * q];
    v[q] = t; s += t; s2 += t * t;
  }
  s  = blockReduceSum(s, sm);
  s2 = blockReduceSum(s2, sm);
  const float mean = s * (1.f / Dc);
  const float rstd = rsqrtf(s2 * (1.f / Dc) - mean * mean + 1e-5f);
#pragma unroll
  for (int q = 0; q < 4; ++q) {
    const int e = threadIdx.x + 256 * q;
    y[off + e] = (v[q] - mean) * rstd * g[e] + b[e];
  }
}

// ------------- K2: causal conv as WMMA GEMM, fused bias + SiLU -------------
// Block = 8 waves sharing (b, 64-wide l strip); wave w owns o-tile (oh*8+w).
// xn row for channel i is staged in LDS (with 64-float causal zero pad) via
// async global->LDS b128 copies, double-buffered across the i loop.
#define CPAD 64
#define CROW (CPAD + Dc)        // 1088 floats per buffer
__global__ __launch_bounds__(256) void conv_wmma_kernel(
    const float* __restrict__ xn, const float* __restrict__ cw,
    const float* __restrict__ cb, float* __restrict__ xc) {
  __shared__ __align__(16) float sx[2][CROW];
  const int tid  = threadIdx.x;
  const int lane = tid & 31;
  const int wave = tid >> 5;
  // blockIdx.x = b*32 + lg*2 + oh   (b in [0,8), lg in [0,16), oh in [0,2))
  const int b  = blockIdx.x >> 5;
  const int lg = (blockIdx.x >> 1) & 15;
  const int oh = blockIdx.x & 1;
  const int o0 = (((oh << 3) + wave) << 4);   // this wave's 16 output channels
  const int l0 = lg << 6;                     // 64-wide output strip base
  const int m  = lane & 15;                   // A row / B col within tile
  const int kb = (lane >> 4) << 1;            // K sub-pair: 0 or 2

  if (tid < CPAD) { sx[0][tid] = 0.f; sx[1][tid] = 0.f; }

  const float* xb = xn + (size_t)b * Sc * Dc;

  // Prologue: async-stage row i=0 into buffer 0 (each thread moves 16B).
  {
    unsigned lo = (unsigned)(size_t)&sx[0][CPAD + tid * 4];
    unsigned long long ga = (unsigned long long)(size_t)(xb + tid * 4);
    asm volatile("global_load_async_to_lds_b128 %0, %1, off"
                 :: "v"(lo), "v"(ga) : "memory");
  }

  v8f acc0 = {}, acc1 = {}, acc2 = {}, acc3 = {};
  const int tmax = l0 + 48 + 12;              // covers t <= l0+63 (strip max)

  for (int i = 0; i < Sc; ++i) {
    asm volatile("s_wait_asynccnt 0x0" ::: "memory");
    __syncthreads();                           // row i visible to all waves
    if (i + 1 < Sc) {                          // prefetch row i+1
      unsigned lo = (unsigned)(size_t)&sx[(i + 1) & 1][CPAD + tid * 4];
      unsigned long long ga =
          (unsigned long long)(size_t)(xb + (size_t)(i + 1) * Dc + tid * 4);
      asm volatile("global_load_async_to_lds_b128 %0, %1, off"
                   :: "v"(lo), "v"(ga) : "memory");
    }
    const float* srow = &sx[i & 1][CPAD];      // srow[j] = xn[b,i,j], zeros j<0
    // Wrev[o0+m, i, t] = cw[o0+m, i, D-1-t]
    const float* wrow = cw + ((size_t)(o0 + m) * Sc + i) * Dc + (Dc - 1);
    for (int t0 = 0; t0 <= tmax; t0 += 4) {
      v2f a;
      a.x = wrow[-(t0 + kb)];
      a.y = wrow[-(t0 + kb + 1)];
      const int jb = l0 + m - t0 - kb;         // n == m; causal pad handles jb<0
      v2f b0, b1, b2, b3;
      b0.x = srow[jb];      b0.y = srow[jb - 1];
      b1.x = srow[jb + 16]; b1.y = srow[jb + 15];
      b2.x = srow[jb + 32]; b2.y = srow[jb + 31];
      b3.x = srow[jb + 48]; b3.y = srow[jb + 47];
      acc0 = wmma4(a, b0, acc0);
      acc1 = wmma4(a, b1, acc1);
      acc2 = wmma4(a, b2, acc2);
      acc3 = wmma4(a, b3, acc3);
    }
  }

  const int half = (lane >> 4) << 3;
  const int n = lane & 15;
#pragma unroll
  for (int r = 0; r < 8; ++r) {
    const int mo = o0 + r + half;
    const float bias = cb[mo];
    float* orow = xc + ((size_t)b * Sc + mo) * Dc + l0 + n;
    float v0 = acc0[r] + bias, v1 = acc1[r] + bias;
    float v2 = acc2[r] + bias, v3 = acc3[r] + bias;
    orow[0]  = v0 / (1.f + expf(-v0));         // SiLU
    orow[16] = v1 / (1.f + expf(-v1));
    orow[32] = v2 / (1.f + expf(-v2));
    orow[48] = v3 / (1.f + expf(-v3));
  }
}

// ------- K3: block-diagonal gate GEMM: out = x*w + h*wr + bias (per blk) ---
__global__ void bd_gemm_kernel(const float* __restrict__ x, const float* __restrict__ w,
                               const float* __restrict__ h, const float* __restrict__ wr,
                               const float* __restrict__ bias, float* __restrict__ out) {
  const int lane = threadIdx.x & 31;
  const int tile = blockIdx.x * (blockDim.x >> 5) + (threadIdx.x >> 5);
  // 128 m-tiles * 4 kblk * 16 n-tiles = 8192
  const int row0 = (tile >> 6) << 4;
  const int kblk = (tile >> 4) & 3;
  const int e0   = (tile & 15) << 4;
  const int m = lane & 15, kb = (lane >> 4) << 1, n = m;
  const int dof = kblk << 8;            // kblk*256

  v8f acc = {};
#pragma unroll
  for (int p = 0; p < 2; ++p) {
    const float* xp = p ? h : x;
    const float* wp = p ? wr : w;
    const float* arow = xp + (size_t)(row0 + m) * Dc + dof;
    const float* wb   = wp + (size_t)kblk * 256 * 256 + e0 + n;
    for (int k0 = 0; k0 < 256; k0 += 4) {
      v2f a;  a.x = arow[k0 + kb];                a.y = arow[k0 + kb + 1];
      v2f bm; bm.x = wb[(size_t)(k0 + kb) * 256]; bm.y = wb[(size_t)(k0 + kb + 1) * 256];
      acc = wmma4(a, bm, acc);
    }
  }
  const int half = (lane >> 4) << 3;
#pragma unroll
  for (int r = 0; r < 8; ++r) {
    const int mo = row0 + r + half;
    out[(size_t)mo * Dc + dof + e0 + n] = acc[r] + bias[dof + e0 + n];
  }
}

// ------ K4: gate nonlinearities + 4 chained LayerNorms (one row/block) -----
__global__ void gates_kernel(const float* __restrict__ iraw, const float* __restrict__ fraw,
                             const float* __restrict__ oraw, const float* __restrict__ zraw,
                             const float* __restrict__ ct1,  const float* __restrict__ nt1,
                             const float* __restrict__ lnc_g, const float* __restrict__ lnc_b,
                             const float* __restrict__ lnn_g, const float* __restrict__ lnn_b,
                             const float* __restrict__ lnh_g, const float* __restrict__ lnh_b,
                             const float* __restrict__ gn_g,  const float* __restrict__ gn_b,
                             float* __restrict__ sout) {
  __shared__ float sm[256];
  const size_t off = (size_t)blockIdx.x * Dc;
  float cv[4], nv[4], ov[4];
  float s1 = 0.f, s2 = 0.f, t1 = 0.f, t2 = 0.f;
#pragma unroll
  for (int q = 0; q < 4; ++q) {
    const int e = threadIdx.x + 256 * q;
    const float iv = expf(iraw[off + e]);
    const float fv = expf(fraw[off + e]);
    const float og = 1.f / (1.f + expf(-oraw[off + e]));
    const float zv = tanhf(zraw[off + e]);
    const float c  = fv * ct1[off + e] + iv * zv;
    const float nn = fv * nt1[off + e] + iv;
    cv[q] = c; nv[q] = nn; ov[q] = og;
    s1 += c; s2 += c * c; t1 += nn; t2 += nn * nn;
  }
  const float cs = blockReduceSum(s1, sm), cs2 = blockReduceSum(s2, sm);
  const float ns = blockReduceSum(t1, sm), ns2 = blockReduceSum(t2, sm);
  const float cm = cs * (1.f / Dc), crs = rsqrtf(cs2 * (1.f / Dc) - cm * cm + 1e-5f);
  const float nm = ns * (1.f / Dc), nrs = rsqrtf(ns2 * (1.f / Dc) - nm * nm + 1e-5f);

  float hv[4], h1 = 0.f, h2 = 0.f;
#pragma unroll
  for (int q = 0; q < 4; ++q) {
    const int e = threadIdx.x + 256 * q;
    const float ct = (cv[q] - cm) * crs * lnc_g[e] + lnc_b[e];
    const float nt = (nv[q] - nm) * nrs * lnn_g[e] + lnn_b[e];
    const float h0 = ov[q] * (ct / nt);
    hv[q] = h0; h1 += h0; h2 += h0 * h0;
  }
  const float hs = blockReduceSum(h1, sm), hs2 = blockReduceSum(h2, sm);
  const float hm = hs * (1.f / Dc), hrs = rsqrtf(hs2 * (1.f / Dc) - hm * hm + 1e-5f);

  float tv[4], g1 = 0.f, g2 = 0.f;
#pragma unroll
  for (int q = 0; q < 4; ++q) {
    const int e = threadIdx.x + 256 * q;
    const float ht = (hv[q] - hm) * hrs * lnh_g[e] + lnh_b[e];
    tv[q] = ht; g1 += ht; g2 += ht * ht;
  }
  const float gs = blockReduceSum(g1, sm), gs2 = blockReduceSum(g2, sm);
  const float gm = gs * (1.f / Dc), grs = rsqrtf(gs2 * (1.f / Dc) - gm * gm + 1e-5f);
#pragma unroll
  for (int q = 0; q < 4; ++q) {
    const int e = threadIdx.x + 256 * q;
    sout[off + e] = (tv[q] - gm) * grs * gn_g[e] + gn_b[e];
  }
}

// ------ K5: g = (s@left_w + lb) * gelu(s@right_w + rb)   [N=1365 padded] ---
__global__ void mlp_in_kernel(const float* __restrict__ s,
                              const float* __restrict__ lw, const float* __restrict__ lb,
                              const float* __restrict__ rw, const float* __restrict__ rb,
                              float* __restrict__ g) {
  const int lane = threadIdx.x & 31;
  const int tile = blockIdx.x * (blockDim.x >> 5) + (threadIdx.x >> 5);
  const int NT = (Hc + 15) / 16;        // 86
  if (tile >= 128 * NT) return;
  const int row0 = (tile / NT) << 4;
  const int col  = ((tile % NT) << 4) + (lane & 15);
  const int m = lane & 15, kb = (lane >> 4) << 1;
  const bool inb = (col < Hc);

  v8f accL = {}, accR = {};
  const float* arow = s + (size_t)(row0 + m) * Dc;
  for (int k0 = 0; k0 < Dc; k0 += 4) {
    v2f a; a.x = arow[k0 + kb]; a.y = arow[k0 + kb + 1];
    v2f bl = {}, br = {};
    if (inb) {
      bl.x = lw[(size_t)(k0 + kb) * Hc + col];
      bl.y = lw[(size_t)(k0 + kb + 1) * Hc + col];
      br.x = rw[(size_t)(k0 + kb) * Hc + col];
      br.y = rw[(size_t)(k0 + kb + 1) * Hc + col];
    }
    accL = wmma4(a, bl, accL);
    accR = wmma4(a, br, accR);
  }
  if (!inb) return;
  const int half = (lane >> 4) << 3;
#pragma unroll
  for (int r = 0; r < 8; ++r) {
    const int mo = row0 + r + half;
    const float l  = accL[r] + lb[col];
    const float rr = accR[r] + rb[col];
    const float ge = 0.5f * rr * (1.f + erff(rr * 0.70710678118654752f)); // exact gelu
    g[(size_t)mo * Hc + col] = l * ge;
  }
}

// ------ K6: out = g @ proj_w + proj_b + xn   [K=1365 zero-padded to 1368] --
__global__ void mlp_out_kernel(const float* __restrict__ g, const float* __restrict__ pw,
                               const float* __restrict__ pb, const float* __restrict__ xn,
                               float* __restrict__ out) {
  const int lane = threadIdx.x & 31;
  const int tile = blockIdx.x * (blockDim.x >> 5) + (threadIdx.x >> 5);
  const int row0 = (tile >> 6) << 4;
  const int n0   = (tile & 63) << 4;
  const int m = lane & 15, kb = (lane >> 4) << 1, n = m;

  v8f acc = {};
  const float* arow = g + (size_t)(row0 + m) * Hc;
  for (int k0 = 0; k0 < 1368; k0 += 4) {
    const int ka = k0 + kb, kc = k0 + kb + 1;
    v2f a, bm;
    a.x  = (ka < Hc) ? arow[ka] : 0.f;
    a.y  = (kc < Hc) ? arow[kc] : 0.f;
    bm.x = (ka < Hc) ? pw[(size_t)ka * Dc + n0 + n] : 0.f;
    bm.y = (kc < Hc) ? pw[(size_t)kc * Dc + n0 + n] : 0.f;
    acc = wmma4(a, bm, acc);
  }
  const int half = (lane >> 4) << 3;
#pragma unroll
  for (int r = 0; r < 8; ++r) {
    const int mo = row0 + r + half;
    const size_t oi = (size_t)mo * Dc + n0 + n;
    out[oi] = acc[r] + pb[n0 + n] + xn[oi];
  }
}

// ---------------------------------------------------------------------------
extern "C" void kernel_launch(void* const* d_in, const int* in_sizes, int n_in,
                              void* d_out, int out_size, void* d_ws, size_t ws_size,
                              hipStream_t stream) {
  (void)in_sizes; (void)n_in; (void)out_size; (void)ws_size;
  const float* x      = (const float*)d_in[0];
  const float* ht_1   = (const float*)d_in[1];
  const float* ct_1   = (const float*)d_in[2];
  const float* nt_1   = (const float*)d_in[3];
  const float* ln_g   = (const float*)d_in[4];
  const float* ln_b   = (const float*)d_in[5];
  const float* conv_w = (const float*)d_in[6];
  const float* conv_b = (const float*)d_in[7];
  const float* i_w    = (const float*)d_in[8];
  const float* i_b    = (const float*)d_in[9];
  const float* f_w    = (const float*)d_in[10];
  const float* f_b    = (const float*)d_in[11];
  const float* o_w    = (const float*)d_in[12];
  const float* o_b    = (const float*)d_in[13];
  const float* z_w    = (const float*)d_in[14];
  const float* z_b    = (const float*)d_in[15];
  const float* ri_w   = (const float*)d_in[16];
  const float* rf_w   = (const float*)d_in[17];
  const float* ro_w   = (const float*)d_in[18];
  const float* rz_w   = (const float*)d_in[19];
  const float* lnc_g  = (const float*)d_in[20];
  const float* lnc_b  = (const float*)d_in[21];
  const float* lnn_g  = (const float*)d_in[22];
  const float* lnn_b  = (const float*)d_in[23];
  const float* lnh_g  = (const float*)d_in[24];
  const float* lnh_b  = (const float*)d_in[25];
  const float* gn_g   = (const float*)d_in[26];
  const float* gn_b   = (const float*)d_in[27];
  const float* left_w = (const float*)d_in[28];
  const float* left_b = (const float*)d_in[29];
  const float* right_w= (const float*)d_in[30];
  const float* right_b= (const float*)d_in[31];
  const float* proj_w = (const float*)d_in[32];
  const float* proj_b = (const float*)d_in[33];

  float* ws = (float*)d_ws;
  const size_t SZ = (size_t)ROWS * Dc;          // 2M floats
  float* xn = ws;
  float* xc = ws + 1 * SZ;
  float* ir = ws + 2 * SZ;
  float* fr = ws + 3 * SZ;
  float* og = ws + 4 * SZ;
  float* zr = ws + 5 * SZ;
  float* sv = ws + 6 * SZ;
  float* gg = ws + 7 * SZ;                      // ROWS*H = 2.80M floats

  ln_kernel<<<ROWS, 256, 0, stream>>>(x, ln_g, ln_b, xn);
  conv_wmma_kernel<<<256, 256, 0, stream>>>(xn, conv_w, conv_b, xc);
  bd_gemm_kernel<<<1024, 256, 0, stream>>>(xc, i_w, ht_1, ri_w, i_b, ir);
  bd_gemm_kernel<<<1024, 256, 0, stream>>>(xc, f_w, ht_1, rf_w, f_b, fr);
  bd_gemm_kernel<<<1024, 256, 0, stream>>>(xn, o_w, ht_1, ro_w, o_b, og);
  bd_gemm_kernel<<<1024, 256, 0, stream>>>(xn, z_w, ht_1, rz_w, z_b, zr);
  gates_kernel<<<ROWS, 256, 0, stream>>>(ir, fr, og, zr, ct_1, nt_1,
                                         lnc_g, lnc_b, lnn_g, lnn_b,
                                         lnh_g, lnh_b, gn_g, gn_b, sv);
  mlp_in_kernel<<<(128 * 86 + 7) / 8, 256, 0, stream>>>(sv, left_w, left_b,
                                                        right_w, right_b, gg);
  mlp_out_kernel<<<1024, 256, 0, stream>>>(gg, proj_w, proj_b, xn, (float*)d_out);
}